// PTPriorNetwork_56813827392360
// MI455X (gfx1250) — compile-verified
//
#include <hip/hip_runtime.h>
#include <hip/hip_bf16.h>

typedef __attribute__((ext_vector_type(2))) float v2f;
typedef __attribute__((ext_vector_type(8))) float v8f;

#define B_ 256
#define N_ 100000
#define D_ 512
#define H_ 2048
#define NB 32                 // table rows per chunk (two 16-col WMMA subtiles)
#define NCHUNK (N_ / NB)      // 3125 exactly

__device__ __forceinline__ unsigned int order_key(float f) {
  // monotone float -> uint mapping (handles negatives)
  unsigned int u = __float_as_uint(f);
  return (u & 0x80000000u) ? ~u : (u | 0x80000000u);
}

__device__ __forceinline__ v8f wmma_f32_k4(v2f a, v2f b, v8f c) {
  return __builtin_amdgcn_wmma_f32_16x16x4_f32(
      false, a, false, b, (short)0, c, false, false);
}

// Async global->LDS copy of 16 bytes (GLOBAL_LOAD_ASYNC_TO_LDS_B128, ASYNCcnt).
// Low 32 bits of a flat shared-aperture address are the LDS byte offset.
__device__ __forceinline__ void async_copy_b128(const void* gsrc, void* ldsdst) {
  unsigned lds = (unsigned)(uintptr_t)ldsdst;
  asm volatile("global_load_async_to_lds_b128 %0, %1, off"
               :: "v"(lds), "v"(gsrc) : "memory");
}

__device__ __forceinline__ void wait_asynccnt_le8() {
  asm volatile("s_wait_asynccnt 0x8" ::: "memory");
}

__global__ void init_min_kernel(unsigned long long* minbuf) {
  int t = threadIdx.x;
  if (t < B_) minbuf[t] = 0xFFFFFFFFFFFFFFFFull;
}

// ---------------------------------------------------------------------------
// Kernel 1: nearest-neighbor argmin.
// Block = 512 threads (16 waves); wave w owns codes rows [16w, 16w+16).
// Double-buffered async staging: prefetch chunk i+1 (32 table rows, 64KB)
// into LDS with global_load_async_to_lds_b128 while the WMMA chain consumes
// chunk i. Each A load (codes) feeds two v_wmma_f32_16x16x4_f32 chains
// (two 16-column subtiles). Running minima packed (order_key<<32|n), reduced
// with 16-lane shfl_xor, merged via global atomic min u64.
// ---------------------------------------------------------------------------
__global__ __launch_bounds__(512)
void nn_argmin_kernel(const float* __restrict__ codes,
                      const float* __restrict__ table,
                      unsigned long long* __restrict__ minbuf) {
  __shared__ float tile[2][NB * D_];   // 2 x 64 KB
  __shared__ float partial[NB * 16];   // 2 KB

  const int t = threadIdx.x;
  const int lane = t & 31;
  const int wave = t >> 5;             // 0..15
  const int mrow = lane & 15;          // M (A) / N (B) within 16
  const int khalf = lane >> 4;         // K half select
  const int m0 = wave * 16;
  const int stride = gridDim.x;

  unsigned long long best[2][8];
#pragma unroll
  for (int nt = 0; nt < 2; ++nt)
#pragma unroll
    for (int r = 0; r < 8; ++r) best[nt][r] = 0xFFFFFFFFFFFFFFFFull;

  const float* aptr = codes + (m0 + mrow) * D_ + 2 * khalf;

  // ---- prologue: stage first chunk into buffer 0 ----
  {
    const char* gsrc = (const char*)table + (size_t)blockIdx.x * (NB * D_ * 4);
    char* ldst = (char*)&tile[0][0];
#pragma unroll
    for (int i = 0; i < 8; ++i) {
      int idx4 = t + i * 512;
      async_copy_b128(gsrc + (size_t)idx4 * 16, ldst + (size_t)idx4 * 16);
    }
  }

  int buf = 0;
  for (int ch = blockIdx.x; ch < NCHUNK; ch += stride) {
    // ---- prefetch next chunk into the other buffer (clamped duplicate
    //      keeps 8 async ops in flight per thread every iteration) ----
    {
      int nch = ch + stride;
      int pch = (nch < NCHUNK) ? nch : ch;
      const char* gsrc = (const char*)table + (size_t)pch * (NB * D_ * 4);
      char* ldst = (char*)&tile[buf ^ 1][0];
#pragma unroll
      for (int i = 0; i < 8; ++i) {
        int idx4 = t + i * 512;
        async_copy_b128(gsrc + (size_t)idx4 * 16, ldst + (size_t)idx4 * 16);
      }
    }
    wait_asynccnt_le8();   // current buffer's 8 copies retired (in-order)
    __syncthreads();       // all waves' slices visible

    const float* tl = &tile[buf][0];

    // ---- per-row sum-of-squares partials (16 partials per row) ----
    {
      int row = t >> 4, sub = t & 15;
      float s = 0.f;
      const float* p = tl + row * D_ + sub * 32;
#pragma unroll
      for (int i = 0; i < 32; ++i) { float v = p[i]; s += v * v; }
      partial[row * 16 + sub] = s;
    }
    __syncthreads();

    float csq0 = 0.f, csq1 = 0.f;
    {
      const float* p0 = &partial[mrow * 16];
      const float* p1 = &partial[(16 + mrow) * 16];
#pragma unroll
      for (int i = 0; i < 16; ++i) { csq0 += p0[i]; csq1 += p1[i]; }
    }

    // ---- two 16x16 dot subtiles, K = 512, shared A loads ----
    v8f acc0 = {0.f, 0.f, 0.f, 0.f, 0.f, 0.f, 0.f, 0.f};
    v8f acc1 = {0.f, 0.f, 0.f, 0.f, 0.f, 0.f, 0.f, 0.f};
    const float* bptr0 = tl + mrow * D_ + 2 * khalf;
    const float* bptr1 = tl + (16 + mrow) * D_ + 2 * khalf;
#pragma unroll 8
    for (int kb = 0; kb < D_; kb += 4) {
      v2f a  = *(const v2f*)(aptr + kb);
      v2f b0 = *(const v2f*)(bptr0 + kb);
      v2f b1 = *(const v2f*)(bptr1 + kb);
      acc0 = wmma_f32_k4(a, b0, acc0);
      acc1 = wmma_f32_k4(a, b1, acc1);
    }

    // ---- fold into running minima (score = csq - 2*dot; monotone in d2) ----
    const unsigned long long n0 = (unsigned long long)(ch * NB + mrow);
#pragma unroll
    for (int r = 0; r < 8; ++r) {
      float s0 = csq0 - 2.0f * acc0[r];
      float s1 = csq1 - 2.0f * acc1[r];
      unsigned long long p0 = ((unsigned long long)order_key(s0) << 32) | n0;
      unsigned long long p1 =
          ((unsigned long long)order_key(s1) << 32) | (n0 + 16ull);
      best[0][r] = (p0 < best[0][r]) ? p0 : best[0][r];
      best[1][r] = (p1 < best[1][r]) ? p1 : best[1][r];
    }
    __syncthreads();   // tile reuse barrier before next prefetch overwrites
    buf ^= 1;
  }

  // ---- reduce across 16 lanes per half-wave, emit global atomic min ----
#pragma unroll
  for (int r = 0; r < 8; ++r) {
    unsigned long long pk =
        (best[0][r] < best[1][r]) ? best[0][r] : best[1][r];
#pragma unroll
    for (int off = 1; off < 16; off <<= 1) {
      unsigned long long o = __shfl_xor(pk, off, 16);
      pk = (o < pk) ? o : pk;
    }
    if (mrow == 0) {
      int m = m0 + r + 8 * khalf;
      atomicMin(&minbuf[m], pk);
    }
  }
}

// ---------------------------------------------------------------------------
// Kernel 2: h1 = relu(table[idx] @ w1 + b1)   (256 x 2048, K = 512)
// One wave per 16x16 output tile; 8 waves per block.
// ---------------------------------------------------------------------------
__global__ __launch_bounds__(256)
void mlp1_kernel(const float* __restrict__ table,
                 const unsigned long long* __restrict__ minbuf,
                 const float* __restrict__ w1,
                 const float* __restrict__ b1,
                 float* __restrict__ h1) {
  const int lane = threadIdx.x & 31;
  const int wave = threadIdx.x >> 5;
  const int mrow = lane & 15, khalf = lane >> 4;
  const int m0 = blockIdx.x * 16;
  const int h0 = (blockIdx.y * 8 + wave) * 16;

  const unsigned int sidx =
      (unsigned int)(minbuf[m0 + mrow] & 0xFFFFFFFFull);
  const float* aptr = table + (size_t)sidx * D_ + 2 * khalf;
  const float* bbase = w1 + (size_t)(2 * khalf) * H_ + (h0 + mrow);

  v8f acc = {0.f, 0.f, 0.f, 0.f, 0.f, 0.f, 0.f, 0.f};
#pragma unroll 8
  for (int kb = 0; kb < D_; kb += 4) {
    v2f a = *(const v2f*)(aptr + kb);
    const float* bp = bbase + (size_t)kb * H_;
    v2f b = {bp[0], bp[H_]};           // w1[k][h], w1[k+1][h]
    acc = wmma_f32_k4(a, b, acc);
  }

  const int h = h0 + mrow;
  const float bias = b1[h];
#pragma unroll
  for (int r = 0; r < 8; ++r) {
    int m = m0 + r + 8 * khalf;
    float v = acc[r] + bias;
    h1[(size_t)m * H_ + h] = v > 0.f ? v : 0.f;
  }
}

// ---------------------------------------------------------------------------
// Kernel 3: mu = h1@w2u + b2u ; logstd = h1@w2s + b2s  (256 x 512, K = 2048)
// Shared A loads feed two WMMA accumulator chains.
// ---------------------------------------------------------------------------
__global__ __launch_bounds__(256)
void mlp2_kernel(const float* __restrict__ h1,
                 const float* __restrict__ w2u, const float* __restrict__ b2u,
                 const float* __restrict__ w2s, const float* __restrict__ b2s,
                 float* __restrict__ out) {
  const int lane = threadIdx.x & 31;
  const int wave = threadIdx.x >> 5;
  const int mrow = lane & 15, khalf = lane >> 4;
  const int m0 = blockIdx.x * 16;
  const int n0 = (blockIdx.y * 8 + wave) * 16;

  const float* aptr = h1 + (size_t)(m0 + mrow) * H_ + 2 * khalf;
  const float* bu = w2u + (size_t)(2 * khalf) * D_ + (n0 + mrow);
  const float* bs = w2s + (size_t)(2 * khalf) * D_ + (n0 + mrow);

  v8f accu = {0.f, 0.f, 0.f, 0.f, 0.f, 0.f, 0.f, 0.f};
  v8f accs = {0.f, 0.f, 0.f, 0.f, 0.f, 0.f, 0.f, 0.f};
#pragma unroll 4
  for (int kb = 0; kb < H_; kb += 4) {
    v2f a = *(const v2f*)(aptr + kb);
    const float* pu = bu + (size_t)kb * D_;
    const float* ps = bs + (size_t)kb * D_;
    v2f vb_u = {pu[0], pu[D_]};
    v2f vb_s = {ps[0], ps[D_]};
    accu = wmma_f32_k4(a, vb_u, accu);
    accs = wmma_f32_k4(a, vb_s, accs);
  }

  const int n = n0 + mrow;
  const float biasu = b2u[n];
  const float biass = b2s[n];
#pragma unroll
  for (int r = 0; r < 8; ++r) {
    int m = m0 + r + 8 * khalf;
    out[(size_t)m * D_ + n] = accu[r] + biasu;                   // mu
    out[(size_t)B_ * D_ + (size_t)m * D_ + n] = accs[r] + biass; // logstd
  }
}

extern "C" void kernel_launch(void* const* d_in, const int* in_sizes, int n_in,
                              void* d_out, int out_size, void* d_ws, size_t ws_size,
                              hipStream_t stream) {
  const float* codes = (const float*)d_in[0];
  const float* table = (const float*)d_in[1];
  const float* w1    = (const float*)d_in[2];
  const float* b1    = (const float*)d_in[3];
  const float* w2u   = (const float*)d_in[4];
  const float* b2u   = (const float*)d_in[5];
  const float* w2s   = (const float*)d_in[6];
  const float* b2s   = (const float*)d_in[7];
  // d_in[8] = k (unused: only neighbor_idx[:,0] feeds the outputs)

  float* out = (float*)d_out;
  unsigned long long* minbuf = (unsigned long long*)d_ws;                 // 2 KB
  float* h1 = (float*)((char*)d_ws + 4096);                               // 2 MB

  init_min_kernel<<<dim3(1), dim3(256), 0, stream>>>(minbuf);
  nn_argmin_kernel<<<dim3(512), dim3(512), 0, stream>>>(codes, table, minbuf);
  mlp1_kernel<<<dim3(16, 16), dim3(256), 0, stream>>>(table, minbuf, w1, b1, h1);
  mlp2_kernel<<<dim3(16, 4), dim3(256), 0, stream>>>(h1, w2u, b2u, w2s, b2s, out);
}